// RTDETRLoss_26920855011631
// MI455X (gfx1250) — compile-verified
//
#include <hip/hip_runtime.h>
#include <hip/hip_bf16.h>
#include <math.h>

#define BB 32
#define QQ 1000
#define NN 100
#define CC 1001      // NUM_CLASSES + 1
#define NOOBJ 1000   // NUM_CLASSES

typedef __attribute__((ext_vector_type(2))) float v2f;
typedef __attribute__((ext_vector_type(8))) float v8f;

// ---------------------------------------------------------------------------
// Wave-level sum via V_WMMA_F32_16X16X4_F32 against an all-ones B matrix.
// A layout (32-bit 16x4): lanes 0-15 hold A[m][0]=s, A[m][1]=0; lanes 16-31
// hold A[m][2]=s, A[m][3]=0.  With B==1 everywhere, D[m][n] = s_m + s_{m+16}.
// Per-lane sum of the 8 C VGPRs gives half the total; one xor-16 shuffle
// completes it.  Requires EXEC all-ones: callers guarantee full waves.
// ---------------------------------------------------------------------------
__device__ __forceinline__ float wave_sum_wmma(float s) {
    v2f a; a[0] = s;    a[1] = 0.0f;
    v2f b; b[0] = 1.0f; b[1] = 1.0f;
    v8f c = {0.f, 0.f, 0.f, 0.f, 0.f, 0.f, 0.f, 0.f};
    c = __builtin_amdgcn_wmma_f32_16x16x4_f32(false, a, false, b,
                                              (short)0, c, false, false);
    float t = c[0] + c[1] + c[2] + c[3] + c[4] + c[5] + c[6] + c[7];
    t += __shfl_xor(t, 16);
    return t;   // full 32-lane sum in every lane
}

// ---------------------------------------------------------------------------
// Kernel 0: zero the loss accumulators (ws is poisoned 0xAA by the harness).
// ---------------------------------------------------------------------------
__global__ void k_zero_acc(float* acc) {
    if (threadIdx.x < 3) acc[threadIdx.x] = 0.0f;
}

// ---------------------------------------------------------------------------
// Kernel 1: per-row max and log-sum-exp over the 1001 logits.
// One wave32 per row, 8 rows per 256-thread block, 4000 blocks exactly.
// Streams the 128 MB logits once (memory-bound phase).
// ---------------------------------------------------------------------------
__global__ void k_row_stats(const float* __restrict__ logits,
                            float* __restrict__ rowmax,
                            float* __restrict__ rowlse) {
    const int wave = threadIdx.x >> 5;
    const int lane = threadIdx.x & 31;
    const int row  = blockIdx.x * 8 + wave;           // always < BB*QQ
    const float* rl = logits + (size_t)row * CC;

    float mx = -3.0e38f;
    for (int k = lane; k < CC; k += 32) {
        if (k + 128 < CC) __builtin_prefetch(rl + k + 128, 0, 0);
        mx = fmaxf(mx, rl[k]);
    }
    for (int off = 16; off > 0; off >>= 1) mx = fmaxf(mx, __shfl_xor(mx, off));

    float s = 0.0f;
    for (int k = lane; k < CC; k += 32) s += expf(rl[k] - mx);
    for (int off = 16; off > 0; off >>= 1) s += __shfl_xor(s, off);

    if (lane == 0) { rowmax[row] = mx; rowlse[row] = logf(s); }
}

// ---------------------------------------------------------------------------
// Kernel 2: cost matrix, stored transposed as (B, N, Q) so the Hungarian
// solver (rows = targets) reads contiguous rows.  Thread owns one q, loops n
// with target boxes/labels staged in LDS; writes are coalesced; the
// prob-gathers hit L2 (logits are L2-resident).
// Faithful to the reference's "buggy-union" pairwise GIoU.
// ---------------------------------------------------------------------------
__global__ void k_cost(const float* __restrict__ logits,
                       const float* __restrict__ pred_boxes,
                       const float* __restrict__ tgt_boxes,
                       const int*   __restrict__ tgt_labels,
                       const float* __restrict__ rowmax,
                       const float* __restrict__ rowlse,
                       float* __restrict__ cost) {
    __shared__ float4 tb[NN];
    __shared__ int    lb[NN];
    const int b = blockIdx.x;
    const int q = blockIdx.y * blockDim.x + threadIdx.x;

    if (threadIdx.x < NN) {
        const float* t = tgt_boxes + ((size_t)b * NN + threadIdx.x) * 4;
        tb[threadIdx.x] = make_float4(t[0], t[1], t[2], t[3]);
        lb[threadIdx.x] = tgt_labels[b * NN + threadIdx.x];
    }
    __syncthreads();
    if (q >= QQ) return;

    const size_t row = (size_t)b * QQ + q;
    const float4 pb  = *(const float4*)(pred_boxes + row * 4);
    const float  m   = rowmax[row];
    const float  l   = rowlse[row];
    const float* lrow = logits + row * CC;

    const float px0 = pb.x - 0.5f * pb.z, py0 = pb.y - 0.5f * pb.w;
    const float px1 = pb.x + 0.5f * pb.z, py1 = pb.y + 0.5f * pb.w;
    const float area1 = (px1 - px0) * (py1 - py0);

    float* crow = cost + (size_t)b * NN * QQ + q;
    for (int n = 0; n < NN; ++n) {
        const float4 t = tb[n];
        const float prob = expf(lrow[lb[n]] - m - l);            // softmax prob
        const float cb = fabsf(pb.x - t.x) + fabsf(pb.y - t.y) +
                         fabsf(pb.z - t.z) + fabsf(pb.w - t.w);  // L1 on cxcywh
        const float tx0 = t.x - 0.5f * t.z, ty0 = t.y - 0.5f * t.w;
        const float tx1 = t.x + 0.5f * t.z, ty1 = t.y + 0.5f * t.w;
        const float area2 = (tx1 - tx0) * (ty1 - ty0);
        float iw = fminf(px1, tx1) - fmaxf(px0, tx0); iw = fmaxf(iw, 0.f);
        float ih = fminf(py1, ty1) - fmaxf(py0, ty0); ih = fmaxf(ih, 0.f);
        const float inter = iw * ih;
        const float iou   = inter / fmaxf(area1 + area2 - inter, 1e-6f);
        float cw = fmaxf(px1, tx1) - fminf(px0, tx0); cw = fmaxf(cw, 0.f);
        float ch = fmaxf(py1, ty1) - fminf(py0, ty0); ch = fmaxf(ch, 0.f);
        const float areac = cw * ch;
        const float unig  = area1 + area2 - iou * area1;          // faithful bug
        const float giou  = iou - (areac - unig) / areac;
        crow[(size_t)n * QQ] = (-prob) + 5.0f * cb - 2.0f * giou;
    }
}

// ---------------------------------------------------------------------------
// Kernel 3: Kuhn-Munkres with potentials, one workgroup per batch (32 run
// concurrently).  n=100 rows (targets), m=1000 cols (queries).  The inner
// min-scan over 1000 columns is parallel across the 256 threads; potentials
// and state live in LDS (~32 KB, fp64 to match the reference solver).
// Writes match[b][t] = matched query index for target t.
// ---------------------------------------------------------------------------
__global__ void k_hungarian(const float* __restrict__ cost, int* __restrict__ match) {
    const int b   = blockIdx.x;
    const int tid = threadIdx.x;
    const float* Cb = cost + (size_t)b * NN * QQ;

    __shared__ double v_[QQ + 1];
    __shared__ double minv[QQ + 1];
    __shared__ double u_[NN + 1];
    __shared__ int    p_[QQ + 1];
    __shared__ int    way[QQ + 1];
    __shared__ int    used[QQ + 1];
    __shared__ double rmin[256];
    __shared__ int    ridx[256];
    __shared__ int    sh_j0, sh_i0, sh_done;
    __shared__ double sh_delta;

    const double INFD = 1.0e300;

    for (int j = tid; j <= QQ; j += 256) { v_[j] = 0.0; p_[j] = 0; }
    if (tid <= NN) u_[tid] = 0.0;
    __syncthreads();

    for (int i = 1; i <= NN; ++i) {
        if (tid == 0) { p_[0] = i; sh_j0 = 0; }
        for (int j = tid; j <= QQ; j += 256) { minv[j] = INFD; used[j] = 0; }
        __syncthreads();

        while (true) {
            if (tid == 0) { used[sh_j0] = 1; sh_i0 = p_[sh_j0]; }
            __syncthreads();
            const int    i0  = sh_i0;
            const int    j0  = sh_j0;
            const double ui0 = u_[i0];

            // relax free columns against row i0
            for (int j = 1 + tid; j <= QQ; j += 256) {
                if (!used[j]) {
                    double cur = (double)Cb[(size_t)(i0 - 1) * QQ + (j - 1)] - ui0 - v_[j];
                    if (cur < minv[j]) { minv[j] = cur; way[j] = j0; }
                }
            }
            __syncthreads();

            // parallel argmin over free columns (ties -> smallest j, like np.argmin)
            double best = INFD; int bj = QQ + 1;
            for (int j = 1 + tid; j <= QQ; j += 256) {
                if (!used[j] && (minv[j] < best || (minv[j] == best && j < bj))) {
                    best = minv[j]; bj = j;
                }
            }
            rmin[tid] = best; ridx[tid] = bj;
            __syncthreads();
            for (int s = 128; s > 0; s >>= 1) {
                if (tid < s) {
                    if (rmin[tid + s] < rmin[tid] ||
                        (rmin[tid + s] == rmin[tid] && ridx[tid + s] < ridx[tid])) {
                        rmin[tid] = rmin[tid + s]; ridx[tid] = ridx[tid + s];
                    }
                }
                __syncthreads();
            }
            if (tid == 0) { sh_delta = rmin[0]; sh_j0 = ridx[0]; }
            __syncthreads();
            const double delta = sh_delta;

            // update potentials (distinct p_[j] per used j -> race-free)
            for (int j = tid; j <= QQ; j += 256) {
                if (used[j]) { u_[p_[j]] += delta; v_[j] -= delta; }
                else         { minv[j]  -= delta; }
            }
            __syncthreads();
            if (tid == 0) sh_done = (p_[sh_j0] == 0);
            __syncthreads();
            if (sh_done) break;
        }

        if (tid == 0) {            // augmenting path
            int j0 = sh_j0;
            while (j0) { int j1 = way[j0]; p_[j0] = p_[j1]; j0 = j1; }
        }
        __syncthreads();
    }

    for (int j = 1 + tid; j <= QQ; j += 256) {
        const int r = p_[j];
        if (r) match[b * NN + (r - 1)] = j - 1;   // target r-1 -> query j-1
    }
}

// ---------------------------------------------------------------------------
// Kernel 4a/4b: build target_classes (default no-object, scatter matched).
// ---------------------------------------------------------------------------
__global__ void k_tc_init(int* tc) {
    const int i = blockIdx.x * blockDim.x + threadIdx.x;
    if (i < BB * QQ) tc[i] = NOOBJ;
}
__global__ void k_tc_scatter(const int* __restrict__ match,
                             const int* __restrict__ labels,
                             int* __restrict__ tc) {
    const int i = blockIdx.x * blockDim.x + threadIdx.x;
    if (i < BB * NN) {
        const int b = i / NN;
        tc[(size_t)b * QQ + match[i]] = labels[i];
    }
}

// ---------------------------------------------------------------------------
// Kernel 5: CE loss sum over all (b,q): -logp[target] = max + lse - logit.
// 125 blocks x 256 threads == 32000 exactly -> every wave is full (EXEC=~0
// requirement of the WMMA reduction holds).
// ---------------------------------------------------------------------------
__global__ void k_ce(const float* __restrict__ logits,
                     const float* __restrict__ rowmax,
                     const float* __restrict__ rowlse,
                     const int*   __restrict__ tc,
                     float* __restrict__ acc) {
    __shared__ float lds[8];
    const int i = blockIdx.x * 256 + threadIdx.x;
    const float term = rowmax[i] + rowlse[i] - logits[(size_t)i * CC + tc[i]];

    const float ws = wave_sum_wmma(term);
    const int lane = threadIdx.x & 31, w = threadIdx.x >> 5;
    if (lane == 0) lds[w] = ws;
    __syncthreads();
    if (threadIdx.x == 0) {
        float r = 0.f;
        for (int k = 0; k < 8; ++k) r += lds[k];
        atomicAdd(&acc[0], r);
    }
}

// ---------------------------------------------------------------------------
// Kernel 6: matched-pair L1 + elementwise GIoU sums.
// 25 blocks x 128 threads == 3200 exactly -> full waves for WMMA reduction.
// ---------------------------------------------------------------------------
__global__ void k_pair(const float* __restrict__ pred_boxes,
                       const float* __restrict__ tgt_boxes,
                       const int*   __restrict__ match,
                       float* __restrict__ acc) {
    __shared__ float lds[4];
    const int i = blockIdx.x * 128 + threadIdx.x;
    const int b = i / NN;
    const int q = match[i];

    const float4 p = *(const float4*)(pred_boxes + ((size_t)b * QQ + q) * 4);
    const float4 g = *(const float4*)(tgt_boxes + (size_t)i * 4);

    const float l1 = fabsf(p.x - g.x) + fabsf(p.y - g.y) +
                     fabsf(p.z - g.z) + fabsf(p.w - g.w);

    const float px0 = p.x - 0.5f * p.z, py0 = p.y - 0.5f * p.w;
    const float px1 = p.x + 0.5f * p.z, py1 = p.y + 0.5f * p.w;
    const float gx0 = g.x - 0.5f * g.z, gy0 = g.y - 0.5f * g.w;
    const float gx1 = g.x + 0.5f * g.z, gy1 = g.y + 0.5f * g.w;
    const float a1 = (px1 - px0) * (py1 - py0);
    const float a2 = (gx1 - gx0) * (gy1 - gy0);
    float iw = fminf(px1, gx1) - fmaxf(px0, gx0); iw = fmaxf(iw, 0.f);
    float ih = fminf(py1, gy1) - fmaxf(py0, gy0); ih = fmaxf(ih, 0.f);
    const float inter = iw * ih;
    const float iou   = inter / fmaxf(a1 + a2 - inter, 1e-6f);
    float cw = fmaxf(px1, gx1) - fminf(px0, gx0); cw = fmaxf(cw, 0.f);
    float ch = fmaxf(py1, gy1) - fminf(py0, gy0); ch = fmaxf(ch, 0.f);
    const float areac = cw * ch;
    const float unig  = a1 + a2 - iou * a1;                 // faithful bug
    const float giou  = iou - (areac - unig) / areac;

    const int lane = threadIdx.x & 31, w = threadIdx.x >> 5;

    const float s1 = wave_sum_wmma(l1);
    if (lane == 0) lds[w] = s1;
    __syncthreads();
    if (threadIdx.x == 0) {
        float r = 0.f; for (int k = 0; k < 4; ++k) r += lds[k];
        atomicAdd(&acc[1], r);
    }
    __syncthreads();
    const float s2 = wave_sum_wmma(giou);
    if (lane == 0) lds[w] = s2;
    __syncthreads();
    if (threadIdx.x == 0) {
        float r = 0.f; for (int k = 0; k < 4; ++k) r += lds[k];
        atomicAdd(&acc[2], r);
    }
}

// ---------------------------------------------------------------------------
// Kernel 7: combine into the scalar loss.
// ---------------------------------------------------------------------------
__global__ void k_final(const float* __restrict__ acc, float* __restrict__ out) {
    const float ce   = acc[0] / (float)(BB * QQ);
    const float bbox = acc[1] / (float)(BB * NN * 4);
    const float giou = ((float)(BB * NN) - acc[2]) / (float)(BB * NN);
    out[0] = ce + 5.0f * bbox + 2.0f * giou;
}

// ---------------------------------------------------------------------------
extern "C" void kernel_launch(void* const* d_in, const int* in_sizes, int n_in,
                              void* d_out, int out_size, void* d_ws, size_t ws_size,
                              hipStream_t stream) {
    const float* logits     = (const float*)d_in[0];   // (B,Q,1001) f32
    const float* pred_boxes = (const float*)d_in[1];   // (B,Q,4)    f32
    const float* tgt_boxes  = (const float*)d_in[2];   // (B,N,4)    f32
    const int*   tgt_labels = (const int*)  d_in[3];   // (B,N)      i32
    float* out = (float*)d_out;

    char*  ws  = (char*)d_ws;
    size_t off = 0;
    auto carve = [&](size_t bytes) -> void* {
        void* p = ws + off;
        off = (off + bytes + 255) & ~(size_t)255;
        return p;
    };
    float* rowmax = (float*)carve((size_t)BB * QQ * 4);
    float* rowlse = (float*)carve((size_t)BB * QQ * 4);
    float* cost   = (float*)carve((size_t)BB * NN * QQ * 4);   // 12.8 MB
    int*   match  = (int*)  carve((size_t)BB * NN * 4);
    int*   tc     = (int*)  carve((size_t)BB * QQ * 4);
    float* acc    = (float*)carve(3 * 4);

    k_zero_acc  <<<1, 32, 0, stream>>>(acc);
    k_row_stats <<<(BB * QQ) / 8, 256, 0, stream>>>(logits, rowmax, rowlse);
    k_cost      <<<dim3(BB, (QQ + 255) / 256), 256, 0, stream>>>(
                    logits, pred_boxes, tgt_boxes, tgt_labels, rowmax, rowlse, cost);
    k_hungarian <<<BB, 256, 0, stream>>>(cost, match);
    k_tc_init   <<<(BB * QQ + 255) / 256, 256, 0, stream>>>(tc);
    k_tc_scatter<<<(BB * NN + 255) / 256, 256, 0, stream>>>(match, tgt_labels, tc);
    k_ce        <<<(BB * QQ) / 256, 256, 0, stream>>>(logits, rowmax, rowlse, tc, acc);
    k_pair      <<<(BB * NN) / 128, 128, 0, stream>>>(pred_boxes, tgt_boxes, match, acc);
    k_final     <<<1, 1, 0, stream>>>(acc, out);
}